// SPPNet_30846455119898
// MI455X (gfx1250) — compile-verified
//
#include <hip/hip_runtime.h>

// SPP patch extraction, (512,48,48) fp32 -> (2828, 25088) fp32.
// Store-bandwidth bound (~284 MB NT stores vs 4.7 MB input).
// CDNA5 path: TENSOR_LOAD_TO_LDS (TDM, TENSORcnt) double-buffered channel
// staging, s_wait_tensorcnt + split-barrier pipeline, separable pooling from
// LDS, non-temporal streaming stores.

typedef unsigned int v4u __attribute__((ext_vector_type(4)));
typedef int          v8i __attribute__((ext_vector_type(8)));
typedef int          v4i __attribute__((ext_vector_type(4)));

#define C_CH   512
#define CPB    8            // channels per block
#define ROWLEN (C_CH * 49)  // 25088 floats per patch row

#define HAVE_TDM __has_builtin(__builtin_amdgcn_tensor_load_to_lds)

// Stage one contiguous band of nelem fp32 (feat channel c, rows y0..y0+s-1)
// into LDS at byte address lds_byte_addr via the Tensor Data Mover.
__device__ __forceinline__ void tdm_stage_band(const float* feat, int c, int y0,
                                               int nelem, unsigned lds_byte_addr) {
#if HAVE_TDM
    unsigned long long ga = (unsigned long long)(feat + (c * 2304 + y0 * 48));
    v4u g0;
    g0[0] = 1u;                                   // count=1: valid user descriptor
    g0[1] = lds_byte_addr;                        // lds_addr (bytes)
    g0[2] = (unsigned)(ga & 0xffffffffu);         // global_addr[31:0]
    g0[3] = (unsigned)((ga >> 32) & 0x01ffffffu)  // global_addr[56:32]
          | 0x80000000u;                          // type=2 ("image")
    v8i g1;
    g1[0] = 0x00020000;     // data_size=2 (4B); no multicast/iterate/pad/barrier
    g1[1] = nelem << 16;    // tensor_dim0[15:0]   (bits 79:48)
    g1[2] = 1 << 16;        // tensor_dim0 hi=0, tensor_dim1[15:0]=1
    g1[3] = nelem << 16;    // tensor_dim1 hi=0, tile_dim0=nelem
    g1[4] = 1;              // tile_dim1=1, tile_dim2=0
    g1[5] = nelem;          // tensor_dim0_stride[31:0]
    g1[6] = 0;
    g1[7] = 0;
    v4i gz = {0, 0, 0, 0};
#if defined(__clang_major__) && (__clang_major__ >= 23)
    v8i gz8 = {0, 0, 0, 0, 0, 0, 0, 0};
    __builtin_amdgcn_tensor_load_to_lds(g0, g1, gz, gz, gz8, 0);
#else
    __builtin_amdgcn_tensor_load_to_lds(g0, g1, gz, gz, 0);
#endif
#endif
}

__device__ __forceinline__ void stage_wait_all_waves() {
#if HAVE_TDM
    __builtin_amdgcn_s_wait_tensorcnt(0);
#else
    asm volatile("s_wait_asynccnt 0x0" ::: "memory");
#endif
}

__global__ __launch_bounds__(256)
void spp_tdm_kernel(const float* __restrict__ feat, float* __restrict__ out) {
    __shared__ float lraw[2][16 * 48];  // staged input rows (double buffered), 6 KB
    __shared__ float lrm[7 * 48];       // row-bin maxima of current channel, 1.3 KB

    const int tid = threadIdx.x;

    // ---- decode (scale band, py) from blockIdx.x in [0, 140) ----
    int bx = blockIdx.x;
    int si = 0, acc = 0, pbase = 0;
    for (int i = 0; i < 7; ++i) {
        int nb = 23 - i;                 // ny == nx for scale s = 4 + 2i
        if (bx < acc + nb) { si = i; break; }
        acc   += nb;
        pbase += nb * nb;
    }
    const int s       = 4 + 2 * si;
    const int n       = 23 - si;
    const int py      = bx - acc;
    const int y0      = py * 2;
    const int rowbase = pbase + py * n;
    const int c0      = blockIdx.y * CPB;
    const int nelem   = s * 48;

    const unsigned ldsraw0 = (unsigned)(uintptr_t)(&lraw[0][0]);
    const unsigned ldsraw1 = (unsigned)(uintptr_t)(&lraw[1][0]);

    // ---- prologue: DMA channel c0 into buffer 0 (TDM ignores EXEC; one issue) ----
    if (tid == 0) {
        tdm_stage_band(feat, c0, y0, nelem, ldsraw0);
    }
#if !HAVE_TDM
    // async fallback: all lanes with tid*16 < nelem*4 stage 16B each
    {
        int goff = (c0 * 2304 + y0 * 48) * 4 + tid * 16;
        if (tid * 16 < nelem * 4) {
            unsigned l = ldsraw0 + (unsigned)(tid * 16);
            asm volatile("global_load_async_to_lds_b128 %0, %1, %2"
                         :: "v"(l), "v"(goff), "s"(feat) : "memory");
        }
    }
#endif

    for (int ci = 0; ci < CPB; ++ci) {
        // Issuing wave's DMA has landed; after the barrier every wave sees the
        // data, and every wave has finished reading lrm and raw[1-(ci&1)].
        stage_wait_all_waves();
        __syncthreads();

        // overlap: DMA next channel into the other raw buffer
        if (ci + 1 < CPB) {
            if (tid == 0) {
                tdm_stage_band(feat, c0 + ci + 1, y0, nelem,
                               ((ci + 1) & 1) ? ldsraw1 : ldsraw0);
            }
#if !HAVE_TDM
            int goff = ((c0 + ci + 1) * 2304 + y0 * 48) * 4 + tid * 16;
            if (tid * 16 < nelem * 4) {
                unsigned l = (((ci + 1) & 1) ? ldsraw1 : ldsraw0) + (unsigned)(tid * 16);
                asm volatile("global_load_async_to_lds_b128 %0, %1, %2"
                             :: "v"(l), "v"(goff), "s"(feat) : "memory");
            }
#endif
        }

        const float* raw = lraw[ci & 1];

        // ---- phase 1: separable pooling, row-bin maxima (7 x 48) ----
        for (int t = tid; t < 7 * 48; t += 256) {
            int bin = t / 48;
            int x   = t - bin * 48;
            // torch adaptive bins: [floor(i*s/7), ceil((i+1)*s/7))
            int ra = (bin * s) / 7, rb = ((bin + 1) * s + 6) / 7;
            float m = raw[ra * 48 + x];
            for (int r = ra + 1; r < rb; ++r) m = fmaxf(m, raw[r * 48 + x]);
            lrm[t] = m;
        }
        __syncthreads();

        // ---- phase 2: column-bin maxima -> outputs ----
        const int c    = c0 + ci;
        const int nout = n * 49;
        int px = tid / 49;
        int k  = tid - px * 49;
        for (int o = tid; o < nout; o += 256) {
            int oy = k / 7;
            int ox = k - oy * 7;
            int ca = (ox * s) / 7, cb = ((ox + 1) * s + 6) / 7;
            const float* rp = lrm + oy * 48 + px * 2;
            float m = rp[ca];
            for (int cc = ca + 1; cc < cb; ++cc) m = fmaxf(m, rp[cc]);

            __builtin_nontemporal_store(
                m, &out[(rowbase + px) * ROWLEN + c * 49 + k]);

            // advance (px,k) by 256 outputs: 256 = 5*49 + 11
            k += 11; px += 5;
            if (k >= 49) { k -= 49; px += 1; }
        }
        // no trailing barrier: next iteration's top barrier covers all hazards
    }
}

extern "C" void kernel_launch(void* const* d_in, const int* in_sizes, int n_in,
                              void* d_out, int out_size, void* d_ws, size_t ws_size,
                              hipStream_t stream) {
    (void)in_sizes; (void)n_in; (void)d_ws; (void)ws_size; (void)out_size;
    const float* feat = (const float*)d_in[0];
    float* out        = (float*)d_out;

    dim3 grid(140, C_CH / CPB);   // 140 (scale, band-row) x 64 channel groups
    dim3 block(256);              // 8 wave32
    spp_tdm_kernel<<<grid, block, 0, stream>>>(feat, out);
}